// NSGP_51419348468001
// MI455X (gfx1250) — compile-verified
//
#include <hip/hip_runtime.h>
#include <math.h>

// ---------------------------------------------------------------------------
// NSGP marginal likelihood for MI455X (gfx1250), fp32 WMMA path.
//   N=3072 points, D=3 dims, M=128 inducing points.
// Heavy ops (k_star@V GEMM, blocked-Cholesky SYRK) use v_wmma_f32_16x16x4_f32
// with 2x2 register blocking (4 independent accumulator chains per wave).
// ---------------------------------------------------------------------------

static constexpr int   NN    = 3072;
static constexpr int   DD    = 3;
static constexpr int   MM    = 128;
static constexpr float TINYF = 1e-20f;
static constexpr float JIT   = 1e-8f;

typedef __attribute__((ext_vector_type(2))) float v2f;
typedef __attribute__((ext_vector_type(8))) float v8f;

#define WMMA_F32(a, b, c) \
    __builtin_amdgcn_wmma_f32_16x16x4_f32(false, (a), false, (b), (short)0, (c), false, false)

// workspace layout (float offsets)
static constexpr size_t OFF_SCAL  = 0;         // 16 scalars: [0]=sumlog(kpost chol) [1]=sumlog(K chol)
static constexpr size_t OFF_YC    = 16;        // N
static constexpr size_t OFF_BVEC  = OFF_YC   + NN;       // N  (rhs / alpha for big solve)
static constexpr size_t OFF_LBUF  = OFF_BVEC + NN;       // D*N  (l[d][i])
static constexpr size_t OFF_ALPH  = OFF_LBUF + (size_t)DD*NN;  // M
static constexpr size_t OFF_KBAR  = OFF_ALPH + MM;       // M*M
static constexpr size_t OFF_KSTAR = OFF_KBAR + (size_t)MM*MM;  // N*M
static constexpr size_t OFF_V     = OFF_KSTAR+ (size_t)NN*MM;  // M*N
static constexpr size_t OFF_BIGA  = OFF_V    + (size_t)MM*NN;  // N*N

// ---------------------------------------------------------------------------
__global__ void zero_scalars(float* s) {
    if (threadIdx.x < 16) s[threadIdx.x] = 0.0f;
}

__global__ void copy_vec(const float* __restrict__ src, float* __restrict__ dst, int n) {
    int i = blockIdx.x * blockDim.x + threadIdx.x;
    if (i < n) dst[i] = src[i];
}

// mean-center y (single workgroup)
__global__ void center_y(const float* __restrict__ y, float* __restrict__ yc) {
    __shared__ float red[1024];
    int t = threadIdx.x;
    float s = 0.f;
    for (int i = t; i < NN; i += 1024) s += y[i];
    red[t] = s; __syncthreads();
    for (int o = 512; o > 0; o >>= 1) { if (t < o) red[t] += red[t + o]; __syncthreads(); }
    float mean = red[0] / (float)NN;
    for (int i = t; i < NN; i += 1024) yc[i] = y[i] - mean;
}

// K_bar = k(xb, xb) + noise^2 I   (M x M)
__global__ void build_Kbar(const float* __restrict__ Xb, const float* __restrict__ ls,
                           const float* __restrict__ stdv, const float* __restrict__ nstd,
                           int d, float* __restrict__ Kb) {
    int idx = blockIdx.x * blockDim.x + threadIdx.x;
    if (idx >= MM * MM) return;
    int i = idx / MM, j = idx % MM;
    float diff = Xb[i * DD + d] - Xb[j * DD + d];
    float dist = diff * diff;
    if (dist == 0.f) dist = TINYF;
    float ls2 = ls[d] * ls[d];
    float v = stdv[d] * stdv[d] * __expf(-0.5f * dist / ls2);
    if (i == j) v += nstd[d] * nstd[d];
    Kb[idx] = v;
}

// k_star = k(X[:,d], Xb[:,d])    (N x M)
__global__ void build_kstar(const float* __restrict__ X, const float* __restrict__ Xb,
                            const float* __restrict__ ls, const float* __restrict__ stdv,
                            int d, float* __restrict__ ks) {
    int idx = blockIdx.x * blockDim.x + threadIdx.x;
    if (idx >= NN * MM) return;
    int i = idx / MM, j = idx % MM;
    float diff = X[i * DD + d] - Xb[j * DD + d];
    float dist = diff * diff;
    if (dist == 0.f) dist = TINYF;
    float ls2 = ls[d] * ls[d];
    ks[idx] = stdv[d] * stdv[d] * __expf(-0.5f * dist / ls2);
}

// In-place lower Cholesky of a 128x128 block of A (leading dim ld) at (off,off).
// Optionally accumulates sum(log(diag(L))) into *acc.
__global__ void chol_block(float* __restrict__ A, size_t ld, int off, float* acc) {
    __shared__ float T[MM * MM];   // 64 KB
    int t = threadIdx.x;           // 128 threads
    for (int r = 0; r < MM; ++r)
        T[r * MM + t] = A[(size_t)(off + r) * ld + off + t];
    __syncthreads();
    for (int k = 0; k < MM; ++k) {
        if (t == k) T[k * MM + k] = sqrtf(fmaxf(T[k * MM + k], 1e-30f));
        __syncthreads();
        float dk = T[k * MM + k];
        if (t > k) T[t * MM + k] /= dk;
        __syncthreads();
        if (t > k) {
            float lik = T[t * MM + k];
            for (int j = k + 1; j <= t; ++j) T[t * MM + j] -= lik * T[j * MM + k];
        }
        __syncthreads();
    }
    for (int r = 0; r < MM; ++r)
        A[(size_t)(off + r) * ld + off + t] = T[r * MM + t];
    if (acc && t == 0) {
        float s = 0.f;
        for (int k = 0; k < MM; ++k) s += logf(fmaxf(T[k * MM + k], 1e-30f));
        atomicAdd(acc, s);
    }
}

// alpha = Kbar^{-1} log|local_ls[:,d]|, Kbar already factored (lower L, ld=M)
__global__ void solve_alpha(const float* __restrict__ L, const float* __restrict__ lls,
                            int d, float* __restrict__ alpha) {
    __shared__ float z[MM];
    int t = threadIdx.x;  // 128
    z[t] = logf(fabsf(lls[t * DD + d]));
    __syncthreads();
    if (t == 0) {
        for (int k = 0; k < MM; ++k) {           // L z' = z
            z[k] /= L[k * MM + k];
            for (int i = k + 1; i < MM; ++i) z[i] -= L[i * MM + k] * z[k];
        }
        for (int k = MM - 1; k >= 0; --k) {      // L^T a = z'
            z[k] /= L[k * MM + k];
            for (int i = 0; i < k; ++i) z[i] -= L[k * MM + i] * z[k];
        }
    }
    __syncthreads();
    alpha[t] = z[t];
}

// l = exp(k_star @ alpha)   (per row dot over M=128)
__global__ void l_gemv(const float* __restrict__ ks, const float* __restrict__ alpha,
                       float* __restrict__ lout) {
    int i = blockIdx.x * blockDim.x + threadIdx.x;
    if (i >= NN) return;
    float s = 0.f;
    for (int j = 0; j < MM; ++j) s += ks[i * MM + j] * alpha[j];
    lout[i] = __expf(s);
}

// V = Kbar^{-1} k_star^T  (M x N); one thread per column, L cached in LDS
__global__ void solve_V(const float* __restrict__ Lg, const float* __restrict__ ks,
                        float* __restrict__ V) {
    __shared__ float L[MM * MM];   // 64 KB
    for (int idx = threadIdx.x; idx < MM * MM; idx += blockDim.x) L[idx] = Lg[idx];
    __syncthreads();
    int c = blockIdx.x * blockDim.x + threadIdx.x;
    if (c >= NN) return;
    for (int k = 0; k < MM; ++k) {               // forward
        float s = ks[(size_t)c * MM + k];
        for (int j = 0; j < k; ++j) s -= L[k * MM + j] * V[(size_t)j * NN + c];
        V[(size_t)k * NN + c] = s / L[k * MM + k];
    }
    for (int k = MM - 1; k >= 0; --k) {          // backward
        float s = V[(size_t)k * NN + c];
        for (int j = k + 1; j < MM; ++j) s -= L[j * MM + k] * V[(size_t)j * NN + c];
        V[(size_t)k * NN + c] = s / L[k * MM + k];
    }
}

// k_post = k(X,X) - k_star @ V, fused epilogue.
// One wave = 32x32 region (2x2 WMMA tiles, 4 independent accumulator chains).
// f32 WMMA frag: lanes0-15 hold K=k0,k0+1 in .x/.y; lanes16-31 hold K=k0+2,k0+3.
__global__ void kpost_wmma(const float* __restrict__ ks, const float* __restrict__ V,
                           const float* __restrict__ X, const float* __restrict__ ls,
                           const float* __restrict__ stdv, int d, float* __restrict__ A) {
    int wave = threadIdx.x >> 5;
    int lane = threadIdx.x & 31;
    int tc = blockIdx.x * 4 + wave;      // 32-wide col block
    int tr = blockIdx.y;                 // 32-tall row block
    int row0 = tr * 32, col0 = tc * 32;
    int hi  = lane >= 16 ? 1 : 0;
    int l16 = lane & 15;
    v8f c00 = {}, c01 = {}, c10 = {}, c11 = {};
    const float* arow0 = ks + (size_t)(row0 + l16) * MM;
    const float* arow1 = ks + (size_t)(row0 + 16 + l16) * MM;
    for (int k0 = 0; k0 < MM; k0 += 4) {
        int ak = k0 + 2 * hi;
        v2f a0, a1, b0, b1;
        a0.x = arow0[ak];     a0.y = arow0[ak + 1];
        a1.x = arow1[ak];     a1.y = arow1[ak + 1];
        b0.x = V[(size_t)ak * NN + col0 + l16];
        b0.y = V[(size_t)(ak + 1) * NN + col0 + l16];
        b1.x = V[(size_t)ak * NN + col0 + 16 + l16];
        b1.y = V[(size_t)(ak + 1) * NN + col0 + 16 + l16];
        c00 = WMMA_F32(a0, b0, c00);
        c01 = WMMA_F32(a0, b1, c01);
        c10 = WMMA_F32(a1, b0, c10);
        c11 = WMMA_F32(a1, b1, c11);
    }
    float ls2 = ls[d] * ls[d];
    float s2  = stdv[d] * stdv[d];
    float xc0 = X[(size_t)(col0 + l16) * DD + d];
    float xc1 = X[(size_t)(col0 + 16 + l16) * DD + d];
    for (int v = 0; v < 8; ++v) {
        int r0 = row0 + v + 8 * hi;
        int r1 = r0 + 16;
        float xr0 = X[(size_t)r0 * DD + d];
        float xr1 = X[(size_t)r1 * DD + d];
        float d00 = (xr0 - xc0) * (xr0 - xc0); if (d00 == 0.f) d00 = TINYF;
        float d01 = (xr0 - xc1) * (xr0 - xc1); if (d01 == 0.f) d01 = TINYF;
        float d10 = (xr1 - xc0) * (xr1 - xc0); if (d10 == 0.f) d10 = TINYF;
        float d11 = (xr1 - xc1) * (xr1 - xc1); if (d11 == 0.f) d11 = TINYF;
        A[(size_t)r0 * NN + col0 + l16]      = s2 * __expf(-0.5f * d00 / ls2) - c00[v];
        A[(size_t)r0 * NN + col0 + 16 + l16] = s2 * __expf(-0.5f * d01 / ls2) - c01[v];
        A[(size_t)r1 * NN + col0 + l16]      = s2 * __expf(-0.5f * d10 / ls2) - c10[v];
        A[(size_t)r1 * NN + col0 + 16 + l16] = s2 * __expf(-0.5f * d11 / ls2) - c11[v];
    }
}

// TRSM: rows below diag block: P <- P * L^{-T}, L block cached in LDS.
__global__ void trsm_panel(float* __restrict__ A, int kb) {
    __shared__ float L[MM * MM];   // 64 KB
    for (int idx = threadIdx.x; idx < MM * MM; idx += blockDim.x)
        L[idx] = A[(size_t)(kb + idx / MM) * NN + kb + (idx % MM)];
    __syncthreads();
    int row = kb + MM + blockIdx.x * blockDim.x + threadIdx.x;
    if (row >= NN) return;
    float* p = A + (size_t)row * NN + kb;
    for (int j = 0; j < MM; ++j) {
        float s = p[j];
        for (int k = 0; k < j; ++k) s -= L[j * MM + k] * p[k];
        p[j] = s / L[j * MM + j];
    }
}

// SYRK trailing update: A[i][j] -= sum_k P[i][k] P[j][k], via WMMA.
// One wave = 32x32 region (2x2 tiles); lower 32x32 blocks only (wave-uniform).
__global__ void syrk_wmma(float* __restrict__ A, int kb) {
    int wave = threadIdx.x >> 5;
    int lane = threadIdx.x & 31;
    int t0 = kb + MM;
    int tc = blockIdx.x * 4 + wave;
    int tr = blockIdx.y;
    int row0 = t0 + tr * 32, col0 = t0 + tc * 32;
    if (col0 >= NN || row0 >= NN) return;
    if (col0 > row0) return;     // lower blocks only (wave-uniform exit)
    int hi  = lane >= 16 ? 1 : 0;
    int l16 = lane & 15;
    v8f c00 = {}, c01 = {}, c10 = {}, c11 = {};
    const float* arow0 = A + (size_t)(row0 + l16) * NN + kb;        // P rows (A frags)
    const float* arow1 = A + (size_t)(row0 + 16 + l16) * NN + kb;
    const float* brow0 = A + (size_t)(col0 + l16) * NN + kb;        // P rows (B = P^T)
    const float* brow1 = A + (size_t)(col0 + 16 + l16) * NN + kb;
    for (int k0 = 0; k0 < MM; k0 += 4) {
        int ak = k0 + 2 * hi;
        v2f a0, a1, b0, b1;
        a0.x = arow0[ak]; a0.y = arow0[ak + 1];
        a1.x = arow1[ak]; a1.y = arow1[ak + 1];
        b0.x = brow0[ak]; b0.y = brow0[ak + 1];
        b1.x = brow1[ak]; b1.y = brow1[ak + 1];
        c00 = WMMA_F32(a0, b0, c00);
        c01 = WMMA_F32(a0, b1, c01);
        c10 = WMMA_F32(a1, b0, c10);
        c11 = WMMA_F32(a1, b1, c11);
    }
    for (int v = 0; v < 8; ++v) {
        int r0 = row0 + v + 8 * hi;
        int r1 = r0 + 16;
        A[(size_t)r0 * NN + col0 + l16]      -= c00[v];
        A[(size_t)r0 * NN + col0 + 16 + l16] -= c01[v];
        A[(size_t)r1 * NN + col0 + l16]      -= c10[v];
        A[(size_t)r1 * NN + col0 + 16 + l16] -= c11[v];
    }
}

// Global kernel K(i,j) from l vectors + X; adds (gnoise^2 + jitter) I.
__global__ void build_K(const float* __restrict__ X, const float* __restrict__ lbuf,
                        const float* __restrict__ gstd, const float* __restrict__ gnoise,
                        float* __restrict__ A) {
    size_t idx = (size_t)blockIdx.x * blockDim.x + threadIdx.x;
    if (idx >= (size_t)NN * NN) return;
    int i = (int)(idx / NN), j = (int)(idx % NN);
    float suffix = 1.f, sdist = 0.f;
    for (int d = 0; d < DD; ++d) {
        float li = lbuf[(size_t)d * NN + i];
        float lj = lbuf[(size_t)d * NN + j];
        float lsq = li * li + lj * lj;
        suffix *= sqrtf(2.f * (li * lj) / lsq);
        float diff = X[(size_t)i * DD + d] - X[(size_t)j * DD + d];
        sdist += diff * diff / lsq;
    }
    float v = gstd[0] * gstd[0] * suffix * __expf(-sdist);
    if (i == j) v += gnoise[0] * gnoise[0] + JIT;
    A[idx] = v;
}

// Forward solve L x = b in place (single workgroup, 1024 threads).
__global__ void fsolve(const float* __restrict__ A, float* __restrict__ b) {
    __shared__ float xk;
    int t = threadIdx.x;
    for (int k = 0; k < NN; ++k) {
        if (t == 0) { xk = b[k] / A[(size_t)k * NN + k]; b[k] = xk; }
        __syncthreads();
        float x = xk;
        for (int i = k + 1 + t; i < NN; i += 1024) b[i] -= A[(size_t)i * NN + k] * x;
        __syncthreads();
    }
}

// Backward solve L^T x = b in place (reads rows of L -> coalesced).
__global__ void bsolve(const float* __restrict__ A, float* __restrict__ b) {
    __shared__ float xk;
    int t = threadIdx.x;
    for (int k = NN - 1; k >= 0; --k) {
        if (t == 0) { xk = b[k] / A[(size_t)k * NN + k]; b[k] = xk; }
        __syncthreads();
        float x = xk;
        for (int i = t; i < k; i += 1024) b[i] -= A[(size_t)k * NN + i] * x;
        __syncthreads();
    }
}

// out = (0.5*(yc.alpha + sumlogdiagL_K) + max(2*sumlogdiagL_kpost, log(TINY))) / N / D
__global__ void finalize(const float* __restrict__ yc, const float* __restrict__ alpha,
                         const float* __restrict__ scal, float* __restrict__ out) {
    __shared__ float red[256];
    int t = threadIdx.x;
    float s = 0.f;
    for (int i = t; i < NN; i += 256) s += yc[i] * alpha[i];
    red[t] = s; __syncthreads();
    for (int o = 128; o > 0; o >>= 1) { if (t < o) red[t] += red[t + o]; __syncthreads(); }
    if (t == 0) {
        float Apart1 = red[0];
        float Apart2 = scal[1];
        float Aval   = 0.5f * (Apart1 + Apart2);
        float Bval   = fmaxf(2.f * scal[0], logf(TINYF));
        out[0] = (Aval + Bval) / (float)NN / (float)DD;
    }
}

// ---------------------------------------------------------------------------
extern "C" void kernel_launch(void* const* d_in, const int* in_sizes, int n_in,
                              void* d_out, int out_size, void* d_ws, size_t ws_size,
                              hipStream_t stream) {
    const float* X      = (const float*)d_in[0];
    const float* y      = (const float*)d_in[1];
    const float* Xb     = (const float*)d_in[2];
    const float* ls     = (const float*)d_in[3];
    const float* stdv   = (const float*)d_in[4];
    const float* nstd   = (const float*)d_in[5];
    const float* lls    = (const float*)d_in[6];
    const float* gstd   = (const float*)d_in[7];
    const float* gnoise = (const float*)d_in[8];
    float* ws = (float*)d_ws;

    float* scal  = ws + OFF_SCAL;
    float* yc    = ws + OFF_YC;
    float* bvec  = ws + OFF_BVEC;
    float* lbuf  = ws + OFF_LBUF;
    float* alphM = ws + OFF_ALPH;
    float* Kbar  = ws + OFF_KBAR;
    float* kstar = ws + OFF_KSTAR;
    float* Vbuf  = ws + OFF_V;
    float* bigA  = ws + OFF_BIGA;

    zero_scalars<<<1, 32, 0, stream>>>(scal);
    center_y<<<1, 1024, 0, stream>>>(y, yc);

    for (int d = 0; d < DD; ++d) {
        build_Kbar<<<(MM * MM + 255) / 256, 256, 0, stream>>>(Xb, ls, stdv, nstd, d, Kbar);
        chol_block<<<1, MM, 0, stream>>>(Kbar, (size_t)MM, 0, nullptr);
        solve_alpha<<<1, MM, 0, stream>>>(Kbar, lls, d, alphM);
        build_kstar<<<(NN * MM + 255) / 256, 256, 0, stream>>>(X, Xb, ls, stdv, d, kstar);
        l_gemv<<<(NN + 255) / 256, 256, 0, stream>>>(kstar, alphM, lbuf + (size_t)d * NN);

        if (d == DD - 1) {   // only last dim's B = log det(k_post) survives in the reference
            solve_V<<<NN / 256, 256, 0, stream>>>(Kbar, kstar, Vbuf);
            // 32x32 per wave, 4 waves (128 cols) per block: grid (N/128, N/32)
            kpost_wmma<<<dim3(NN / 128, NN / 32), 128, 0, stream>>>(kstar, Vbuf, X, ls,
                                                                    stdv, d, bigA);
            for (int kb = 0; kb < NN; kb += MM) {   // blocked Cholesky, log-det accum
                chol_block<<<1, MM, 0, stream>>>(bigA, (size_t)NN, kb, &scal[0]);
                int rows = NN - kb - MM;
                if (rows > 0) {
                    trsm_panel<<<(rows + 255) / 256, 256, 0, stream>>>(bigA, kb);
                    int nb32 = rows / 32;
                    syrk_wmma<<<dim3((nb32 + 3) / 4, nb32), 128, 0, stream>>>(bigA, kb);
                }
            }
        }
    }

    // Global GP: K, Cholesky, alpha = K^{-1} yc
    build_K<<<(int)(((size_t)NN * NN + 255) / 256), 256, 0, stream>>>(X, lbuf, gstd,
                                                                      gnoise, bigA);
    for (int kb = 0; kb < NN; kb += MM) {
        chol_block<<<1, MM, 0, stream>>>(bigA, (size_t)NN, kb, &scal[1]);
        int rows = NN - kb - MM;
        if (rows > 0) {
            trsm_panel<<<(rows + 255) / 256, 256, 0, stream>>>(bigA, kb);
            int nb32 = rows / 32;
            syrk_wmma<<<dim3((nb32 + 3) / 4, nb32), 128, 0, stream>>>(bigA, kb);
        }
    }
    copy_vec<<<(NN + 255) / 256, 256, 0, stream>>>(yc, bvec, NN);
    fsolve<<<1, 1024, 0, stream>>>(bigA, bvec);
    bsolve<<<1, 1024, 0, stream>>>(bigA, bvec);
    finalize<<<1, 256, 0, stream>>>(yc, bvec, scal, (float*)d_out);

    (void)in_sizes; (void)n_in; (void)out_size; (void)ws_size;
}